// SelfAttention_77343771066393
// MI455X (gfx1250) — compile-verified
//
#include <hip/hip_runtime.h>

// ---------------------------------------------------------------------------
// Self-attention (B=8, C=256, K=64, H=W=64) for gfx1250 / MI455X.
// All matrix math on V_WMMA_F32_16X16X4_F32 (fp32 needed: logits std~8,
// exp() amplifies bf16 logit error ~10%). 3-pass, never materializes the
// 537MB attn matrix:
//   pass1: Q=[Wq@X], K=[Wk@X], V=[Wv@X]  (raw reshape == reinterpret)
//   pass2: per-row m[i], 1/Z[i] of S=Q@K^T (online, WMMA tiles)
//   pass3: out[:,j] = sum_i (V[:,i]/Z[i]) * exp(S[i,j]-m[i]) + x  (fused
//          two-level WMMA: S-tile -> exp -> LDS -> second GEMM)
// ---------------------------------------------------------------------------

typedef __attribute__((ext_vector_type(2))) float v2f;
typedef __attribute__((ext_vector_type(4))) float v4f;
typedef __attribute__((ext_vector_type(8))) float v8f;

__device__ __forceinline__ v8f wmma_f32(v2f a, v2f b, v8f c) {
  // D = A(16x4) * B(4x16) + C(16x16), fp32
  return __builtin_amdgcn_wmma_f32_16x16x4_f32(
      false, a, false, b, (short)0, c, false, false);
}

#define HWC 4096   // H*W
#define CCH 256    // C
#define KCH 64     // K

// ---------------------------------------------------------------------------
// Pass 1: Out[b][m][s] = sum_c Wm[m][c] * X[b][c][s] + bias[m]
// 64x64 output tile per WG, 8 waves, k-chunks of 16 staged in LDS.
// ---------------------------------------------------------------------------
__global__ __launch_bounds__(256)
void qkv_proj(const float* __restrict__ X, const float* __restrict__ Wm,
              const float* __restrict__ bias, float* __restrict__ Out, int M) {
  const int b   = blockIdx.z;
  const int n0  = blockIdx.x * 64;
  const int m0  = blockIdx.y * 64;
  const int tid = threadIdx.x;
  const int lane = tid & 31, wv = tid >> 5;
  const int l15 = lane & 15, hi = lane >> 4;

  __shared__ __align__(16) float Xs[16 * 64];  // [k][n]
  __shared__ __align__(16) float Ws[64 * 16];  // [m][k]

  const float* Xb = X + (size_t)b * CCH * HWC;

  const int mt  = wv >> 1;        // 0..3  (16-row tile)
  const int nt0 = (wv & 1) * 2;   // 0 or 2 (two 16-col tiles)

  v8f acc0 = {}; v8f acc1 = {};

  for (int kc = 0; kc < CCH; kc += 16) {
    __syncthreads();
    {  // stage X[kc:kc+16, n0:n0+64] (coalesced float4)
      int idx = tid * 4;                 // 0..1023
      int r = idx >> 6, c = idx & 63;
      *(v4f*)&Xs[idx] = *(const v4f*)&Xb[(size_t)(kc + r) * HWC + n0 + c];
    }
    {  // stage W[m0:m0+64, kc:kc+16]
      int idx = tid * 4;
      int m = idx >> 4, k = idx & 15;
      *(v4f*)&Ws[idx] = *(const v4f*)&Wm[(size_t)(m0 + m) * CCH + kc + k];
    }
    __syncthreads();
#pragma unroll
    for (int s = 0; s < 4; ++s) {
      const int t0 = s * 4;
      v2f a;
      a.x = Ws[(mt * 16 + l15) * 16 + t0 + 2 * hi];
      a.y = Ws[(mt * 16 + l15) * 16 + t0 + 2 * hi + 1];
      v2f b0, b1;
      b0.x = Xs[(t0 + 2 * hi) * 64 + nt0 * 16 + l15];
      b0.y = Xs[(t0 + 2 * hi + 1) * 64 + nt0 * 16 + l15];
      b1.x = Xs[(t0 + 2 * hi) * 64 + (nt0 + 1) * 16 + l15];
      b1.y = Xs[(t0 + 2 * hi + 1) * 64 + (nt0 + 1) * 16 + l15];
      acc0 = wmma_f32(a, b0, acc0);
      acc1 = wmma_f32(a, b1, acc1);
    }
  }

  float* Ob = Out + (size_t)b * M * HWC;
#pragma unroll
  for (int r = 0; r < 8; ++r) {
    int row = m0 + mt * 16 + r + 8 * hi;   // C/D layout: vgpr r, halves +8
    float bs = bias[row];
    Ob[(size_t)row * HWC + n0 + nt0 * 16 + l15]       = acc0[r] + bs;
    Ob[(size_t)row * HWC + n0 + (nt0 + 1) * 16 + l15] = acc1[r] + bs;
  }
}

// ---------------------------------------------------------------------------
// Pass 2: per-row max m[i] and 1/Z[i] over S[i,:] = Q[i,:] . K[j,:].
// One wave owns 16 rows; Q-row fragments live in registers for all 256
// j-steps; per-lane online max/sum, single cross-lane merge at the end.
// ---------------------------------------------------------------------------
__global__ __launch_bounds__(128)
void softmax_stats(const float* __restrict__ Q, const float* __restrict__ Kb,
                   float* __restrict__ Mrow, float* __restrict__ IZrow) {
  const int b = blockIdx.z;
  const int lane = threadIdx.x & 31, wv = threadIdx.x >> 5;
  const int l15 = lane & 15, hi = lane >> 4;
  const int i0 = blockIdx.x * 64 + wv * 16;

  const float* Qb = Q + (size_t)b * KCH * HWC;   // [4096][64] reinterpreted
  const float* Kp = Kb + (size_t)b * KCH * HWC;

  v2f aq[16];
  {
    const float* qrow = Qb + (size_t)(i0 + l15) * 64;
#pragma unroll
    for (int s = 0; s < 16; ++s) {
      aq[s].x = qrow[4 * s + 2 * hi];
      aq[s].y = qrow[4 * s + 2 * hi + 1];
    }
  }

  float m[8], ssum[8];
#pragma unroll
  for (int r = 0; r < 8; ++r) { m[r] = -3.0e38f; ssum[r] = 0.0f; }

  for (int j0 = 0; j0 < HWC; j0 += 16) {
    const float* krow = Kp + (size_t)(j0 + l15) * 64;
    v8f acc = {};
#pragma unroll
    for (int s = 0; s < 16; ++s) {
      v2f bk;
      bk.x = krow[4 * s + 2 * hi];
      bk.y = krow[4 * s + 2 * hi + 1];
      acc = wmma_f32(aq[s], bk, acc);
    }
#pragma unroll
    for (int r = 0; r < 8; ++r) {   // per-lane online max/sum (no shuffles)
      float x  = acc[r];
      float nm = fmaxf(m[r], x);
      ssum[r]  = ssum[r] * __expf(m[r] - nm) + __expf(x - nm);
      m[r] = nm;
    }
  }
  // merge the 16 column-slices of each row (xor<16 stays inside half-wave)
#pragma unroll
  for (int r = 0; r < 8; ++r) {
    float mm = m[r], ss = ssum[r];
#pragma unroll
    for (int d = 1; d < 16; d <<= 1) {
      float mo = __shfl_xor(mm, d, 32);
      float so = __shfl_xor(ss, d, 32);
      float nm = fmaxf(mm, mo);
      ss = ss * __expf(mm - nm) + so * __expf(mo - nm);
      mm = nm;
    }
    m[r] = mm; ssum[r] = ss;
  }
  if (l15 == 0) {
#pragma unroll
    for (int r = 0; r < 8; ++r) {
      int row = i0 + r + 8 * hi;
      Mrow[(size_t)b * HWC + row]  = m[r];
      IZrow[(size_t)b * HWC + row] = 1.0f / ssum[r];
    }
  }
}

// ---------------------------------------------------------------------------
// Pass 3: out[b][c][j] = sum_i (V[c,i]/Z[i]) * exp(S[i,j]-m[i]) + x[b][c][j]
// One WG per 64-column block: K-tile resident in LDS; each i-iteration
// stages Q-tile, computes S via WMMA, writes E=exp(S-m) back over the Q
// LDS region, then a second WMMA level accumulates V'@E.
// ---------------------------------------------------------------------------
__global__ __launch_bounds__(256)
void attn_out(const float* __restrict__ Q, const float* __restrict__ Kb,
              const float* __restrict__ V, const float* __restrict__ Mrow,
              const float* __restrict__ IZ, const float* __restrict__ X,
              float* __restrict__ Out) {
  const int b  = blockIdx.z;
  const int j0 = blockIdx.x * 64;
  const int tid = threadIdx.x;
  const int lane = tid & 31, wv = tid >> 5;
  const int l15 = lane & 15, hi = lane >> 4;

  __shared__ __align__(16) float Ks[64 * 64];  // K rows j0..j0+63  (16 KB)
  __shared__ __align__(16) float QE[64 * 64];  // Q tile, then E    (16 KB)
  __shared__ float msh[64];
  __shared__ float izsh[64];

  const float* Qb = Q  + (size_t)b * KCH * HWC;
  const float* Kp = Kb + (size_t)b * KCH * HWC;
  const float* Vb = V  + (size_t)b * CCH * HWC;
  const float* Mb = Mrow + (size_t)b * HWC;
  const float* Zb = IZ   + (size_t)b * HWC;

  // K rows j0..j0+63 are contiguous: flat 16KB copy
  for (int idx = tid * 4; idx < 64 * 64; idx += 256 * 4)
    *(v4f*)&Ks[idx] = *(const v4f*)&Kp[(size_t)j0 * 64 + idx];

  v8f acc[2][4];
#pragma unroll
  for (int ct = 0; ct < 2; ++ct)
#pragma unroll
    for (int jt = 0; jt < 4; ++jt) acc[ct][jt] = (v8f){};

  const int c0  = wv * 32;          // each wave owns 32 C-rows
  const int it  = wv >> 1;          // S tile assignment
  const int jt0 = (wv & 1) * 2;

  for (int i0 = 0; i0 < HWC; i0 += 64) {
    __syncthreads();
    for (int idx = tid * 4; idx < 64 * 64; idx += 1024)   // stage Q tile
      *(v4f*)&QE[idx] = *(const v4f*)&Qb[(size_t)i0 * 64 + idx];
    if (tid < 64) { msh[tid] = Mb[i0 + tid]; izsh[tid] = Zb[i0 + tid]; }
    if (i0 + 64 < HWC)                                    // warm next tile
      __builtin_prefetch(&Qb[(size_t)(i0 + 64) * 64], 0, 1);
    __syncthreads();

    // ---- level 1: S tile (this wave: 1 i-tile x 2 j-tiles) ----
    v8f s0 = {}; v8f s1 = {};
#pragma unroll
    for (int s = 0; s < 16; ++s) {
      v2f a;
      a.x = QE[(it * 16 + l15) * 64 + 4 * s + 2 * hi];
      a.y = QE[(it * 16 + l15) * 64 + 4 * s + 2 * hi + 1];
      v2f b0, b1;
      b0.x = Ks[(jt0 * 16 + l15) * 64 + 4 * s + 2 * hi];
      b0.y = Ks[(jt0 * 16 + l15) * 64 + 4 * s + 2 * hi + 1];
      b1.x = Ks[((jt0 + 1) * 16 + l15) * 64 + 4 * s + 2 * hi];
      b1.y = Ks[((jt0 + 1) * 16 + l15) * 64 + 4 * s + 2 * hi + 1];
      s0 = wmma_f32(a, b0, s0);
      s1 = wmma_f32(a, b1, s1);
    }
    __syncthreads();   // all waves done reading Q tile

    // ---- E = exp(S - m[i]) into the Q LDS region ----
#pragma unroll
    for (int r = 0; r < 8; ++r) {
      int il = it * 16 + r + 8 * hi;
      float mr = msh[il];
      QE[il * 64 + jt0 * 16 + l15]       = __expf(s0[r] - mr);
      QE[il * 64 + (jt0 + 1) * 16 + l15] = __expf(s1[r] - mr);
    }
    __syncthreads();

    // ---- level 2: acc += (V[c0:c0+32, i0:i0+64] * 1/Z) @ E ----
#pragma unroll
    for (int s = 0; s < 16; ++s) {
      const float iz0 = izsh[4 * s + 2 * hi];
      const float iz1 = izsh[4 * s + 2 * hi + 1];
      v2f a0, a1;
      a0.x = Vb[(size_t)(c0 + l15) * HWC + i0 + 4 * s + 2 * hi] * iz0;
      a0.y = Vb[(size_t)(c0 + l15) * HWC + i0 + 4 * s + 2 * hi + 1] * iz1;
      a1.x = Vb[(size_t)(c0 + 16 + l15) * HWC + i0 + 4 * s + 2 * hi] * iz0;
      a1.y = Vb[(size_t)(c0 + 16 + l15) * HWC + i0 + 4 * s + 2 * hi + 1] * iz1;
#pragma unroll
      for (int jt = 0; jt < 4; ++jt) {
        v2f bb;
        bb.x = QE[(4 * s + 2 * hi) * 64 + jt * 16 + l15];
        bb.y = QE[(4 * s + 2 * hi + 1) * 64 + jt * 16 + l15];
        acc[0][jt] = wmma_f32(a0, bb, acc[0][jt]);
        acc[1][jt] = wmma_f32(a1, bb, acc[1][jt]);
      }
    }
  }

  // ---- epilogue: residual add + store ----
  const float* Xb = X + (size_t)b * CCH * HWC;
  float* Ob = Out + (size_t)b * CCH * HWC;
#pragma unroll
  for (int ct = 0; ct < 2; ++ct)
#pragma unroll
    for (int jt = 0; jt < 4; ++jt)
#pragma unroll
      for (int r = 0; r < 8; ++r) {
        int c = c0 + ct * 16 + r + 8 * hi;
        int j = j0 + jt * 16 + l15;
        Ob[(size_t)c * HWC + j] = acc[ct][jt][r] + Xb[(size_t)c * HWC + j];
      }
}

// ---------------------------------------------------------------------------
extern "C" void kernel_launch(void* const* d_in, const int* in_sizes, int n_in,
                              void* d_out, int out_size, void* d_ws,
                              size_t ws_size, hipStream_t stream) {
  (void)in_sizes; (void)n_in; (void)out_size; (void)ws_size;
  const float* x  = (const float*)d_in[0];
  const float* Wq = (const float*)d_in[1];
  const float* bq = (const float*)d_in[2];
  const float* Wk = (const float*)d_in[3];
  const float* bk = (const float*)d_in[4];
  const float* Wv = (const float*)d_in[5];
  const float* bv = (const float*)d_in[6];

  // workspace carve-out (~50.6 MB)
  float* qb = (float*)d_ws;                       // 8*64*4096
  float* kb = qb + (size_t)8 * KCH * HWC;         // 8*64*4096
  float* vb = kb + (size_t)8 * KCH * HWC;         // 8*256*4096
  float* mr = vb + (size_t)8 * CCH * HWC;         // 8*4096
  float* iz = mr + (size_t)8 * HWC;               // 8*4096

  qkv_proj<<<dim3(HWC / 64, 1, 8), 256, 0, stream>>>(x, Wq, bq, qb, KCH);
  qkv_proj<<<dim3(HWC / 64, 1, 8), 256, 0, stream>>>(x, Wk, bk, kb, KCH);
  qkv_proj<<<dim3(HWC / 64, CCH / 64, 8), 256, 0, stream>>>(x, Wv, bv, vb, CCH);
  softmax_stats<<<dim3(HWC / 64, 1, 8), 128, 0, stream>>>(qb, kb, mr, iz);
  attn_out<<<dim3(HWC / 64, 1, 8), 256, 0, stream>>>(qb, kb, vb, mr, iz, x,
                                                     (float*)d_out);
}